// GraphConvolution2_82179904241989
// MI455X (gfx1250) — compile-verified
//
#include <hip/hip_runtime.h>

// ---------------------------------------------------------------------------
// GCN layer for MI455X (gfx1250, wave32):
//   out = segment_sum(x[edge_col] * edge_val) @ W + b
// restructured as  y = x @ W  (dense fp32 WMMA GEMM),
//                  out[n] = sum_{e in row n} val[e] * y[col[e]] + b
// ---------------------------------------------------------------------------

#define GCN_N 100000
#define GCN_D 128

typedef __attribute__((ext_vector_type(2))) float v2f;
typedef __attribute__((ext_vector_type(8))) float v8f;

// ---------------------------------------------------------------------------
// Kernel 1: y = x @ W.   x:[N,128] row-major, W:[128,128] row-major.
// Block = 256 threads = 8 waves; block covers 128 rows; wave covers 16 rows
// x 128 cols as 8 accumulator tiles of 16x16 fp32 (v8f each).
// K-loop: 32 steps of V_WMMA_F32_16X16X4_F32 (full fp32 precision).
// ---------------------------------------------------------------------------
__global__ __launch_bounds__(256) void gcn_gemm_xw_kernel(
    const float* __restrict__ x, const float* __restrict__ w,
    float* __restrict__ y, int nRows)
{
    // Full W in LDS: 128x128 fp32 = 64 KB (WGP has 320 KB).
    __shared__ float wlds[GCN_D * GCN_D];

    // Coalesced cooperative load of W.
    for (int i = threadIdx.x; i < GCN_D * GCN_D; i += 256)
        wlds[i] = w[i];
    __syncthreads();

    const int wave = threadIdx.x >> 5;
    const int lane = threadIdx.x & 31;
    const int m    = lane & 15;          // M index within 16x16 tile
    const int kh   = (lane >> 4) << 1;   // K sub-pair: lanes 0-15 -> K{0,1}, 16-31 -> K{2,3}
    const int row0 = blockIdx.x * 128 + wave * 16;

    // Clamp row for safe loads on the tail block; stores are guarded below.
    int rowA = row0 + m;
    if (rowA >= nRows) rowA = nRows - 1;
    const float* __restrict__ xrow = x + (size_t)rowA * GCN_D;

    v8f acc[8] = {};

    for (int k = 0; k < GCN_D; k += 4) {
        // A fragment (16x4 fp32): lane holds x[row0+m][k+kh .. k+kh+1]
        v2f a = *(const v2f*)(xrow + k + kh);
#pragma unroll
        for (int t = 0; t < 8; ++t) {
            // B fragment (4x16 fp32): lane holds W[k+kh][n], W[k+kh+1][n], n = t*16 + m
            const int n = (t << 4) + m;
            v2f b;
            b.x = wlds[(k + kh) * GCN_D + n];
            b.y = wlds[(k + kh + 1) * GCN_D + n];
            acc[t] = __builtin_amdgcn_wmma_f32_16x16x4_f32(
                /*neg_a=*/false, a, /*neg_b=*/false, b,
                /*c_mod=*/(short)0, acc[t],
                /*reuse_a=*/false, /*reuse_b=*/false);
        }
    }

    // C/D layout: VGPR r -> (M=r, N=lane) for lanes 0-15, (M=r+8, N=lane-16) for 16-31.
    const int mhi = (lane >> 4) << 3;
#pragma unroll
    for (int t = 0; t < 8; ++t) {
        const int n = (t << 4) + m;
#pragma unroll
        for (int r = 0; r < 8; ++r) {
            const int rr = row0 + mhi + r;
            if (rr < nRows)
                y[(size_t)rr * GCN_D + n] = acc[t][r];
        }
    }
}

// ---------------------------------------------------------------------------
// Kernel 2: out[node] = sum_{e: edge_row[e]==node} edge_val[e] * y[edge_col[e]] + bias
// edge_row is sorted -> each node's edges are contiguous; one wave per node
// binary-searches its [beg,end) range. No atomics, fully deterministic.
// Edge metadata goes through scalar (SMEM) loads: range is readfirstlane'd
// into SGPRs so ec[e]/ev[e] are wave-uniform addresses.
// Each lane owns 4 of the 128 output dims (float4).
// ---------------------------------------------------------------------------
__global__ __launch_bounds__(256) void gcn_spmm_bias_kernel(
    const float* __restrict__ y,
    const int*   __restrict__ er,
    const int*   __restrict__ ec,
    const float* __restrict__ ev,
    const float* __restrict__ bias,
    float* __restrict__ out, int nE)
{
    const int lane = threadIdx.x & 31;
    const int node = blockIdx.x * 8 + (threadIdx.x >> 5);
    if (node >= GCN_N) return;

    // lower_bound(er, node)
    int lo = 0, hi = nE;
    while (lo < hi) {
        int mid = (lo + hi) >> 1;
        if (er[mid] < node) lo = mid + 1; else hi = mid;
    }
    const int beg = __builtin_amdgcn_readfirstlane(lo);
    // upper_bound(er, node), continuing from lo
    hi = nE;
    while (lo < hi) {
        int mid = (lo + hi) >> 1;
        if (er[mid] <= node) lo = mid + 1; else hi = mid;
    }
    const int end = __builtin_amdgcn_readfirstlane(lo);

    float4 acc = make_float4(0.f, 0.f, 0.f, 0.f);
    const int d4 = lane << 2;

    for (int e = beg; e < end; ++e) {
        const int   col = ec[e];           // scalar load (uniform e)
        const float val = ev[e];           // scalar load (uniform e)
        const float4 v = *(const float4*)(y + (size_t)col * GCN_D + d4);
        acc.x = fmaf(v.x, val, acc.x);
        acc.y = fmaf(v.y, val, acc.y);
        acc.z = fmaf(v.z, val, acc.z);
        acc.w = fmaf(v.w, val, acc.w);
    }

    const float4 b = *(const float4*)(bias + d4);
    float4 r;
    r.x = acc.x + b.x;
    r.y = acc.y + b.y;
    r.z = acc.z + b.z;
    r.w = acc.w + b.w;
    *(float4*)(out + (size_t)node * GCN_D + d4) = r;
}

// ---------------------------------------------------------------------------
// Inputs (setup_inputs order): x, edge_row, edge_col, edge_val, weight, bias
// d_ws holds y = x@W : 100000*128*4 = 51.2 MB.
// ---------------------------------------------------------------------------
extern "C" void kernel_launch(void* const* d_in, const int* in_sizes, int n_in,
                              void* d_out, int out_size, void* d_ws, size_t ws_size,
                              hipStream_t stream)
{
    const float* x    = (const float*)d_in[0];
    const int*   er   = (const int*)  d_in[1];
    const int*   ec   = (const int*)  d_in[2];
    const float* ev   = (const float*)d_in[3];
    const float* w    = (const float*)d_in[4];
    const float* bias = (const float*)d_in[5];
    float* out = (float*)d_out;
    float* y   = (float*)d_ws;

    const int nE    = in_sizes[1];
    const int nRows = in_sizes[0] / GCN_D;

    dim3 blk(256);
    dim3 g1((nRows + 127) / 128);
    gcn_gemm_xw_kernel<<<g1, blk, 0, stream>>>(x, w, y, nRows);

    dim3 g2((nRows + 7) / 8);
    gcn_spmm_bias_kernel<<<g2, blk, 0, stream>>>(y, er, ec, ev, bias, out, nE);
}